// RoIHeads_19902878450311
// MI455X (gfx1250) — compile-verified
//
#include <hip/hip_runtime.h>
#include <hip/hip_bf16.h>

// Problem dims (from reference setup_inputs)
#define BB 16
#define PP 8000
#define GG 64
#define CC 91
#define NN (BB * PP)          // 128000 rows
#define ROWS_PER_BLK 8        // 8 wave32 per 256-thread block
#define NB2 (NN / ROWS_PER_BLK)

static_assert(NN % ROWS_PER_BLK == 0, "rows divisible");

typedef __attribute__((ext_vector_type(2))) float v2f;
typedef __attribute__((ext_vector_type(8))) float v8f;

__device__ __forceinline__ float wave_sum(float x) {
  #pragma unroll
  for (int m = 1; m <= 16; m <<= 1) x += __shfl_xor(x, m, 32);
  return x;
}
__device__ __forceinline__ float wave_max(float x) {
  #pragma unroll
  for (int m = 1; m <= 16; m <<= 1) x = fmaxf(x, __shfl_xor(x, m, 32));
  return x;
}

// 32-lane sum using V_WMMA_F32_16X16X4_F32:
// A = ones(16x4), B: vgpr0 = x, vgpr1 = 0  ->  D[m,n] = x[n] + x[n+16]
// (holds for either {K0,K2}/{K1,K3} or {K0,K1}/{K2,K3} VGPR row packing).
// d[0] per lane = x[lane&15] + x[(lane&15)+16]; finish with 4 xor-shuffles.
__device__ __forceinline__ float wmma_sum32(float x) {
#if defined(__gfx1250__) && __has_builtin(__builtin_amdgcn_wmma_f32_16x16x4_f32)
  v2f a; a[0] = 1.0f; a[1] = 1.0f;
  v2f b; b[0] = x;    b[1] = 0.0f;
  v8f c = {};
  c = __builtin_amdgcn_wmma_f32_16x16x4_f32(false, a, false, b, (short)0, c,
                                            false, false);
  float y = c[0];
#else
  float y = x + __shfl_xor(x, 16, 32);
#endif
  y += __shfl_xor(y, 1, 32);
  y += __shfl_xor(y, 2, 32);
  y += __shfl_xor(y, 4, 32);
  y += __shfl_xor(y, 8, 32);
  return y;
}

// ---------------- Kernel 1: proposal <-> GT matching + box encode ----------
__global__ __launch_bounds__(256) void k_match(
    const float* __restrict__ prop,   // [B,P,4]
    const float* __restrict__ gtb,    // [B,G,4]
    const int*   __restrict__ gtl,    // [B,G]
    int*    __restrict__ labels,      // [N]
    float4* __restrict__ regt) {      // [N]
  __shared__ float s_box[GG * 4];
  __shared__ int   s_lab[GG];
  const int b   = blockIdx.y;
  const int tid = threadIdx.x;
  for (int i = tid; i < GG * 4; i += 256) s_box[i] = gtb[(size_t)b * GG * 4 + i];
  for (int i = tid; i < GG; i += 256)     s_lab[i] = gtl[(size_t)b * GG + i];
  __syncthreads();

  const int p = blockIdx.x * 256 + tid;
  if (p >= PP) return;
  const size_t row = (size_t)b * PP + p;
  const float4 pb = reinterpret_cast<const float4*>(prop)[row];
  const float pw = pb.z - pb.x, ph = pb.w - pb.y;
  const float area_p = pw * ph;

  float best = -1.0f; int bi = 0;
  #pragma unroll 4
  for (int g = 0; g < GG; ++g) {
    const float ax = s_box[4 * g + 0], ay = s_box[4 * g + 1];
    const float ax2 = s_box[4 * g + 2], ay2 = s_box[4 * g + 3];
    const float area_g = (ax2 - ax) * (ay2 - ay);
    const float iw = fmaxf(fminf(ax2, pb.z) - fmaxf(ax, pb.x), 0.0f);
    const float ih = fmaxf(fminf(ay2, pb.w) - fmaxf(ay, pb.y), 0.0f);
    const float inter = iw * ih;
    const float iou = inter / (area_g + area_p - inter);
    if (iou > best) { best = iou; bi = g; }   // strict '>' == first argmax
  }
  int lab = s_lab[bi];
  if (best < 0.5f) lab = 0;   // BG_TH == FG_TH == 0.5 -> no "ignore" band

  // encode(gt[bi], proposal) with weights (10,10,5,5)
  const float gx = s_box[4 * bi + 0], gy = s_box[4 * bi + 1];
  const float gx2 = s_box[4 * bi + 2], gy2 = s_box[4 * bi + 3];
  const float gw = gx2 - gx, gh = gy2 - gy;
  const float ecx = pb.x + 0.5f * pw, ecy = pb.y + 0.5f * ph;
  const float gcx = gx + 0.5f * gw,   gcy = gy + 0.5f * gh;
  float4 t;
  t.x = 10.0f * (gcx - ecx) / pw;
  t.y = 10.0f * (gcy - ecy) / ph;
  t.z = 5.0f * logf(gw / pw);
  t.w = 5.0f * logf(gh / ph);
  labels[row] = lab;
  regt[row] = t;
}

// ---------------- Kernel 2: per-row CE + smooth-L1, block partial sums -----
__global__ __launch_bounds__(256) void k_loss(
    const float*  __restrict__ cls,    // [N,C]
    const float*  __restrict__ breg,   // [N,C*4]
    const int*    __restrict__ labels, // [N]
    const float4* __restrict__ regt,   // [N]
    float4*       __restrict__ partials) {
  const int wave = threadIdx.x >> 5;
  const int lane = threadIdx.x & 31;
  const size_t row = (size_t)blockIdx.x * ROWS_PER_BLK + wave;

  float nll = 0.0f, vld = 0.0f, bx = 0.0f;
  if (row < NN) {
    const float* rp = cls + row * (size_t)CC;
    const int lab  = labels[row];
    const int safe = (lab >= 0) ? lab : 0;
    const float* gptr = breg + row * (size_t)(CC * 4) + (size_t)safe * 4;
    if (lane == 0) __builtin_prefetch(gptr, 0, 0);  // hide dependent gather

    // 91 logits across 32 lanes: slots lane, lane+32, lane+64
    const float v0 = rp[lane];
    const float v1 = rp[lane + 32];
    const float v2 = (lane + 64 < CC) ? rp[lane + 64] : -__builtin_inff();

    const float m = wave_max(fmaxf(v0, fmaxf(v1, v2)));
    float s = expf(v0 - m) + expf(v1 - m);
    if (lane + 64 < CC) s += expf(v2 - m);
    s = wave_sum(s);
    const float lse = m + logf(s);

    if (lane == 0) {
      vld = 1.0f;                 // FG_TH==BG_TH -> no ignored rows
      nll = lse - rp[safe];       // -log p[safe]
      if (lab > 0) {
        const float4 bs = *reinterpret_cast<const float4*>(gptr); // 16B aligned
        const float4 rt = regt[row];
        const float kBeta = 1.0f / 9.0f;
        const float d0 = fabsf(bs.x - rt.x), d1 = fabsf(bs.y - rt.y);
        const float d2 = fabsf(bs.z - rt.z), d3 = fabsf(bs.w - rt.w);
        bx  = (d0 < kBeta) ? 4.5f * d0 * d0 : d0 - 0.5f * kBeta;
        bx += (d1 < kBeta) ? 4.5f * d1 * d1 : d1 - 0.5f * kBeta;
        bx += (d2 < kBeta) ? 4.5f * d2 * d2 : d2 - 0.5f * kBeta;
        bx += (d3 < kBeta) ? 4.5f * d3 * d3 : d3 - 0.5f * kBeta;
      }
    }
  }

  __shared__ float red[ROWS_PER_BLK][4];
  if (lane == 0) { red[wave][0] = nll; red[wave][1] = vld; red[wave][2] = bx; }
  __syncthreads();
  if (threadIdx.x == 0) {
    float a = 0.0f, b = 0.0f, c = 0.0f;
    #pragma unroll
    for (int w = 0; w < ROWS_PER_BLK; ++w) {  // fixed order -> deterministic
      a += red[w][0]; b += red[w][1]; c += red[w][2];
    }
    partials[blockIdx.x] = make_float4(a, b, c, 0.0f);
  }
}

// ---------------- Kernel 3: deterministic final reduce (WMMA 32-lane sum) --
__global__ __launch_bounds__(32) void k_final(
    const float4* __restrict__ partials, int nb, float* __restrict__ out) {
  const int lane = threadIdx.x;
  float s_nll = 0.0f, s_cnt = 0.0f, s_box = 0.0f;
  for (int i = lane; i < nb; i += 32) {     // fixed lane->chunk map
    const float4 v = partials[i];
    s_nll += v.x; s_cnt += v.y; s_box += v.z;
  }
  s_nll = wmma_sum32(s_nll);   // v_wmma_f32_16x16x4_f32 (ones x B) reduce
  s_cnt = wmma_sum32(s_cnt);
  s_box = wmma_sum32(s_box);
  if (lane == 0) {
    out[0] = s_nll / fmaxf(s_cnt, 1.0f);
    out[1] = s_box * (1.0f / (float)NN);
  }
}

extern "C" void kernel_launch(void* const* d_in, const int* in_sizes, int n_in,
                              void* d_out, int out_size, void* d_ws, size_t ws_size,
                              hipStream_t stream) {
  const float* cls  = (const float*)d_in[0];   // [N,C]
  const float* breg = (const float*)d_in[1];   // [N,C*4]
  const float* prop = (const float*)d_in[2];   // [B,P,4]
  const float* gtb  = (const float*)d_in[3];   // [B,G,4]
  const int*   gtl  = (const int*)d_in[4];     // [B,G]
  float* out = (float*)d_out;

  // Scratch layout (all 16B aligned): labels | reg targets | block partials
  char* w = (char*)d_ws;
  int*    labels   = (int*)w;                               // N * 4B
  float4* regt     = (float4*)(w + (size_t)NN * 4);         // N * 16B
  float4* partials = (float4*)(w + (size_t)NN * 4 + (size_t)NN * 16); // NB2*16B

  k_match<<<dim3((PP + 255) / 256, BB), 256, 0, stream>>>(prop, gtb, gtl,
                                                          labels, regt);
  k_loss<<<NB2, 256, 0, stream>>>(cls, breg, labels, regt, partials);
  k_final<<<1, 32, 0, stream>>>(partials, NB2, out);
}